// WindowAttention_40578851012963
// MI455X (gfx1250) — compile-verified
//
#include <hip/hip_runtime.h>
#include <math.h>

// ---------------------------------------------------------------------------
// Window attention for MI455X (gfx1250, wave32, WMMA).
// All matrix math uses V_WMMA_F32_16X16X4_F32 (full fp32): the problem is
// bandwidth/latency bound (~2.7 GFLOP, ~25MB unique traffic @ 23.3 TB/s HBM,
// 12MB kqv intermediate lives in the 192MB L2), so reference-matching fp32
// precision costs nothing at the roofline.
// GEMMs use 32x32-per-wave register tiling (4 WMMAs / 4 frag loads per k-step)
// plus near-scope global_prefetch_b8 one K-chunk ahead to hide L2 latency.
// ---------------------------------------------------------------------------

typedef __attribute__((ext_vector_type(2))) float v2f;
typedef __attribute__((ext_vector_type(8))) float v8f;

#define WMMA_F32(a, b, c) \
  __builtin_amdgcn_wmma_f32_16x16x4_f32(false, (a), false, (b), (short)0, (c), false, false)

constexpr int B_ = 2, T_ = 2048, C_ = 256, H_ = 4, D_ = 64, W_ = 64, S_ = 1;
constexpr int Tq_ = T_ - S_;            // 2047
constexpr int KQVROW = 3 * H_ * D_;     // 768

__device__ __forceinline__ int clampi(int v, int lo, int hi) {
  return v < lo ? lo : (v > hi ? hi : v);
}

// ---------------------------------------------------------------------------
// C[M,N] = A[M,K] @ B[N,K]^T (+ bias[N]).  One wave -> one 32x32 tile:
// 2 A-frags x 2 B-frags -> 4 WMMAs per k-step, halving frag-load traffic
// vs a 16x16 tile.  M, N must be multiples of 32; K multiple of 64.
// ---------------------------------------------------------------------------
__global__ __launch_bounds__(32)
void gemm_abt(const float* __restrict__ A, const float* __restrict__ Bm,
              const float* __restrict__ bias, float* __restrict__ Cm,
              int M, int N, int K) {
  const int lane = threadIdx.x & 31;
  const int half = lane >> 4;       // 0 or 1
  const int l16  = lane & 15;
  const int kb   = half * 2;        // A/B frag K sub-offset {0,2}
  const int n0   = blockIdx.x * 32;
  const int m0   = blockIdx.y * 32;

  const float* a0 = A  + (size_t)(m0 + l16)      * K;   // A frag: M = lane&15
  const float* a1 = A  + (size_t)(m0 + 16 + l16) * K;
  const float* b0 = Bm + (size_t)(n0 + l16)      * K;   // B frag: N = lane&15
  const float* b1 = Bm + (size_t)(n0 + 16 + l16) * K;

  v8f c00 = {}, c01 = {}, c10 = {}, c11 = {};
  for (int k0 = 0; k0 < K; k0 += 64) {
    // near-scope speculative prefetch of the next 64-float K-chunk per stream
    __builtin_prefetch(a0 + k0 + 64, 0, 3);
    __builtin_prefetch(a1 + k0 + 64, 0, 3);
    __builtin_prefetch(b0 + k0 + 64, 0, 3);
    __builtin_prefetch(b1 + k0 + 64, 0, 3);
#pragma unroll 4
    for (int k = k0; k < k0 + 64; k += 4) {
      v2f fa0 = *(const v2f*)(a0 + k + kb);
      v2f fa1 = *(const v2f*)(a1 + k + kb);
      v2f fb0 = *(const v2f*)(b0 + k + kb);
      v2f fb1 = *(const v2f*)(b1 + k + kb);
      c00 = WMMA_F32(fa0, fb0, c00);
      c01 = WMMA_F32(fa0, fb1, c01);
      c10 = WMMA_F32(fa1, fb0, c10);
      c11 = WMMA_F32(fa1, fb1, c11);
    }
  }
  const float bb0 = bias ? bias[n0 + l16]      : 0.0f;
  const float bb1 = bias ? bias[n0 + 16 + l16] : 0.0f;
#pragma unroll
  for (int i = 0; i < 8; ++i) {
    const int m = m0 + i + half * 8;                  // C/D layout
    Cm[(size_t)m * N        + (n0 + l16)]      = c00[i] + bb0;
    Cm[(size_t)m * N        + (n0 + 16 + l16)] = c01[i] + bb1;
    Cm[(size_t)(m + 16) * N + (n0 + l16)]      = c10[i] + bb0;
    Cm[(size_t)(m + 16) * N + (n0 + 16 + l16)] = c11[i] + bb1;
  }
}

// ---------------------------------------------------------------------------
// s0[(b*H+h)*T + j] = dot(k[b,j,h,:], q[b,j,h,:])
// ---------------------------------------------------------------------------
__global__ __launch_bounds__(256)
void s0_kernel(const float* __restrict__ kqv, float* __restrict__ s0) {
  const int idx = blockIdx.x * 256 + threadIdx.x;     // (b*H+h)*T + j
  if (idx >= B_ * H_ * T_) return;
  const int j  = idx % T_;
  const int bh = idx / T_;
  const int b  = bh / H_, h = bh % H_;
  const float* base = kqv + (size_t)(b * T_ + j) * KQVROW + h * D_;
  const float* kp = base;                 // slot 0 = k
  const float* qp = base + H_ * D_;       // slot 1 = q
  float acc = 0.0f;
#pragma unroll 8
  for (int d = 0; d < D_; ++d) acc = fmaf(kp[d], qp[d], acc);
  s0[idx] = acc;
}

// ---------------------------------------------------------------------------
// S1[b,h,j,w] = dot(q[b,j,h,:], q_pos[w,h,:])   (WMMA, per (b,h): 2048x64x64)
// ---------------------------------------------------------------------------
__global__ __launch_bounds__(32)
void s1_kernel(const float* __restrict__ kqv, const float* __restrict__ q_pos,
               float* __restrict__ S1) {
  const int lane = threadIdx.x & 31;
  const int half = lane >> 4, l16 = lane & 15, kb = half * 2;
  const int j0 = blockIdx.x * 16;
  const int w0 = blockIdx.y * 16;
  const int bh = blockIdx.z;
  const int b = bh / H_, h = bh % H_;

  const float* qrow = kqv + (size_t)(b * T_ + j0 + l16) * KQVROW + H_ * D_ + h * D_;
  const float* prow = q_pos + (size_t)((w0 + l16) * H_ + h) * D_;

  v8f c = {};
#pragma unroll
  for (int k = 0; k < D_; k += 4) {
    v2f a = *(const v2f*)(qrow + k + kb);
    v2f b2 = *(const v2f*)(prow + k + kb);
    c = WMMA_F32(a, b2, c);
  }
  float* out = S1 + ((size_t)bh * T_ + j0) * W_ + w0;
#pragma unroll
  for (int i = 0; i < 8; ++i)
    out[(size_t)(i + half * 8) * W_ + l16] = c[i];
}

// ---------------------------------------------------------------------------
// Zero the t=0 (pad) rows of attnout (rest are written by attn_kernel).
// ---------------------------------------------------------------------------
__global__ __launch_bounds__(256)
void zero_pad_rows(float* __restrict__ attnout) {
  const int i = threadIdx.x;
  attnout[i] = 0.0f;                      // b = 0, row 0
  attnout[(size_t)T_ * C_ + i] = 0.0f;    // b = 1, row 0
}

// ---------------------------------------------------------------------------
// Per (b, h, 16-t tile): assemble scores, softmax over w, emit wei, then
// out = P_banded @ V (5 j-tiles, WMMA) + wei @ v_pos (WMMA).
// 128 threads = 4 waves; wave w owns d-chunk [16w, 16w+16).
// ---------------------------------------------------------------------------
__global__ __launch_bounds__(128)
void attn_kernel(const float* __restrict__ kqv, const float* __restrict__ S1,
                 const float* __restrict__ s0, const float* __restrict__ v_pos,
                 float* __restrict__ attnout, float* __restrict__ wei_out) {
  __shared__ float sc[16][W_ + 1];        // +1 pad: avoid 64-bank conflicts

  const int t0 = blockIdx.x * 16;
  const int h  = blockIdx.y;
  const int b  = blockIdx.z;
  const int bh = b * H_ + h;
  const int tid = threadIdx.x;
  const float scale = 0.125f;             // D^-0.5 = 1/8

  // ---- scores: each of 128 threads fills 8 entries of the 16x64 tile ----
  {
    const int ti = tid >> 3;
    const int wb = (tid & 7) * 8;
    const int t  = t0 + ti;
#pragma unroll
    for (int u = 0; u < 8; ++u) {
      const int w = wb + u;
      const int j = t + w - (W_ - 1);     // absolute key index
      float v;
      if (t >= Tq_)      v = 0.0f;        // dead row; keep softmax finite
      else if (j < 0)    v = -INFINITY;   // reference mask
      else               v = (s0[(size_t)bh * T_ + j] +
                              S1[((size_t)bh * T_ + j) * W_ + w]) * scale;
      sc[ti][w] = v;
    }
  }
  __syncthreads();

  // ---- softmax over w (one thread per row) ----
  if (tid < 16) {
    float m = -INFINITY;
    for (int w = 0; w < W_; ++w) m = fmaxf(m, sc[tid][w]);
    float s = 0.0f;
    for (int w = 0; w < W_; ++w) { float e = __expf(sc[tid][w] - m); sc[tid][w] = e; s += e; }
    const float inv = 1.0f / s;
    for (int w = 0; w < W_; ++w) sc[tid][w] *= inv;
  }
  __syncthreads();

  // ---- write wei output ----
  {
    const int ti = tid >> 3;
    const int wb = (tid & 7) * 8;
    const int t  = t0 + ti;
    if (t < Tq_) {
#pragma unroll
      for (int u = 0; u < 8; ++u) {
        const int w = wb + u;
        wei_out[(((size_t)b * Tq_ + t) * W_ + w) * H_ + h] = sc[ti][w];
      }
    }
  }

  // ---- output GEMMs (per wave d-chunk), EXEC full for all WMMA ----
  const int lane = tid & 31;
  const int wave = tid >> 5;
  const int d0   = wave * 16;
  const int half = lane >> 4, l16 = lane & 15, kb = half * 2;
  const float* vbase = kqv + 2 * H_ * D_ + h * D_ + d0 + l16;

  v8f c = {};
  // out1: banded sum_j P[t,j] * v[j,:], j in [t0-63, t0+15] -> 5 aligned tiles
#pragma unroll
  for (int jt = 0; jt < 5; ++jt) {
    const int j0 = t0 - 64 + jt * 16;
#pragma unroll
    for (int k4 = 0; k4 < 16; k4 += 4) {
      // A fragment: P[l16][k4+kb .. +1], P[t][jj] = wei[t][j-t+63] (0 if OOB)
      v2f a;
      {
        const int t = t0 + l16;
        const int wa = (j0 + k4 + kb) - t + (W_ - 1);
        const int wA = clampi(wa, 0, W_ - 1);
        const int wB = clampi(wa + 1, 0, W_ - 1);
        a.x = (wa     >= 0 && wa     < W_) ? sc[l16][wA] : 0.0f;
        a.y = (wa + 1 >= 0 && wa + 1 < W_) ? sc[l16][wB] : 0.0f;
      }
      // B fragment: v[j0+k4+kb+{0,1}][d0+l16] (clamped addr; A is 0 when OOB)
      v2f bv;
      {
        const int jA = clampi(j0 + k4 + kb,     0, T_ - 1);
        const int jB = clampi(j0 + k4 + kb + 1, 0, T_ - 1);
        bv.x = vbase[(size_t)(b * T_ + jA) * KQVROW];
        bv.y = vbase[(size_t)(b * T_ + jB) * KQVROW];
      }
      c = WMMA_F32(a, bv, c);
    }
  }
  // out2: wei(16x64) @ v_pos[:,h,d-chunk](64x16)
#pragma unroll
  for (int k = 0; k < W_; k += 4) {
    v2f a;
    a.x = sc[l16][k + kb];
    a.y = sc[l16][k + kb + 1];
    v2f bv;
    bv.x = v_pos[((size_t)(k + kb)     * H_ + h) * D_ + d0 + l16];
    bv.y = v_pos[((size_t)(k + kb + 1) * H_ + h) * D_ + d0 + l16];
    c = WMMA_F32(a, bv, c);
  }

  // store: attnout[b, t+S, h*D + d]  (S=1 time shift from reference pad)
#pragma unroll
  for (int i = 0; i < 8; ++i) {
    const int ti = i + half * 8;
    const int t  = t0 + ti;
    if (t < Tq_)
      attnout[((size_t)b * T_ + (t + S_)) * C_ + h * D_ + d0 + l16] = c[i];
  }
}

// ---------------------------------------------------------------------------
extern "C" void kernel_launch(void* const* d_in, const int* in_sizes, int n_in,
                              void* d_out, int out_size, void* d_ws, size_t ws_size,
                              hipStream_t stream) {
  (void)in_sizes; (void)n_in; (void)out_size; (void)ws_size;

  const float* x     = (const float*)d_in[0];
  const float* w_kqv = (const float*)d_in[1];
  const float* w_out = (const float*)d_in[2];
  const float* b_out = (const float*)d_in[3];
  const float* q_pos = (const float*)d_in[4];
  const float* v_pos = (const float*)d_in[5];

  float* y   = (float*)d_out;                       // (B, T, C)
  float* wei = y + (size_t)B_ * T_ * C_;            // (B, Tq, W, H)

  // workspace layout
  char*  ws      = (char*)d_ws;
  float* kqv     = (float*)ws;                                  // 4096 x 768
  size_t off     = (size_t)B_ * T_ * KQVROW * sizeof(float);
  float* S1      = (float*)(ws + off);                          // (B*H, T, W)
  off           += (size_t)B_ * H_ * T_ * W_ * sizeof(float);
  float* s0      = (float*)(ws + off);                          // (B*H, T)
  off           += (size_t)B_ * H_ * T_ * sizeof(float);
  float* attnout = (float*)(ws + off);                          // 4096 x 256

  // 1) kqv = x @ w_kqv^T            (M=4096, N=768, K=256)
  gemm_abt<<<dim3(KQVROW / 32, (B_ * T_) / 32), 32, 0, stream>>>(
      x, w_kqv, nullptr, kqv, B_ * T_, KQVROW, C_);

  // 2) s0 (k.q dots) and S1 (q @ q_pos^T per (b,h))
  s0_kernel<<<(B_ * H_ * T_ + 255) / 256, 256, 0, stream>>>(kqv, s0);
  s1_kernel<<<dim3(T_ / 16, W_ / 16, B_ * H_), 32, 0, stream>>>(kqv, q_pos, S1);

  // 3) zero the S=1 pad rows of attnout
  zero_pad_rows<<<1, C_, 0, stream>>>(attnout);

  // 4) windowed softmax + banded/positional output GEMMs
  attn_kernel<<<dim3((Tq_ + 15) / 16, H_, B_), 128, 0, stream>>>(
      kqv, S1, s0, v_pos, attnout, wei);

  // 5) y = attnout @ w_out^T + b_out   (M=4096, N=256, K=256)
  gemm_abt<<<dim3(C_ / 32, (B_ * T_) / 32), 32, 0, stream>>>(
      attnout, w_out, b_out, y, B_ * T_, C_, C_);
}